// Net_67714454389272
// MI455X (gfx1250) — compile-verified
//
#include <hip/hip_runtime.h>
#include <hip/hip_bf16.h>
#include <stddef.h>

// ---------------------------------------------------------------------------
// Projected-LSTM scan fused end-to-end for MI455X (gfx1250, wave32).
//
// B=4096, T=2048, P=3, H=32, 4H=128, K=128.  Memory is trivial (~106 MB @
// 23.3 TB/s); the kernel is a latency/VALU-bound sequential recurrence, so:
//  * one wave per batch element -> 4096 independent waves (full occupancy);
//  * lane j owns cell state c[j] and gate rows {j, 32+j, 64+j, 96+j};
//  * the recurrence-independent input projection X . W_ih^T is computed 16
//    timesteps at a time with 8x V_WMMA_F32_16X16X4_F32 (K=4 >= P=3, exact
//    f32), software-pipelined one block ahead of the sequential consumer;
//  * ALL cross-lane traffic is on the VALU pipe: v_permlane16_swap_b32 for
//    WMMA-tile steering (one swap serves steps v and v+8 simultaneously) and
//    DPP16 (quad_perm / row_ror) + one swap for the h-feedback reduction.
//    Zero LDS ops -> no s_wait_dscnt on the sequential chain;
//  * activations use v_tanh_f32 (gfx1250 trans op): 1 trans + 1 fma each;
//  * A-operand load is branchless (per-lane offset + cndmask), avoiding
//    exec-mask save/restore and double waits on the per-block seam.
// ---------------------------------------------------------------------------

#define LSTM_B 4096
#define LSTM_T 2048
#define LSTM_P 3
#define LSTM_H 32
#define LSTM_K 128

typedef __attribute__((ext_vector_type(2))) float v2f;
typedef __attribute__((ext_vector_type(8))) float v8f;

// ---- cross-lane primitives (VALU only) ------------------------------------

// v_permlane16_swap_b32: swaps lanes 16..31 of x with lanes 0..15 of y.
__device__ __forceinline__ void half_swap(float& x, float& y) {
    int a = __float_as_int(x), b = __float_as_int(y);
    asm("v_permlane16_swap_b32 %0, %1" : "+v"(a), "+v"(b));
    x = __int_as_float(a);
    y = __int_as_float(b);
}

// v += dpp_permute(v) within 16-lane rows (DPP16 control as template ICE).
template <int CTRL>
__device__ __forceinline__ float dpp_xadd(float v) {
    int t = __builtin_amdgcn_update_dpp(0, __float_as_int(v), CTRL,
                                        0xf, 0xf, true);
    return v + __int_as_float(t);
}

// Full 32-lane sum, result in every lane. 4 DPP adds + 1 permlane swap.
__device__ __forceinline__ float wave_allsum(float v) {
    v = dpp_xadd<0xB1>(v);   // quad_perm [1,0,3,2]  : xor 1
    v = dpp_xadd<0x4E>(v);   // quad_perm [2,3,0,1]  : xor 2
    v = dpp_xadd<0x124>(v);  // row_ror:4
    v = dpp_xadd<0x128>(v);  // row_ror:8  -> row sum in all 16 lanes
    float a = v, b = v;
    half_swap(a, b);         // a={lo,lo}, b={hi,hi}
    return a + b;            // full sum in all 32 lanes
}

// ---- activations (v_tanh_f32 confirmed on gfx1250) -------------------------

__device__ __forceinline__ float fast_tanh(float x) {
#if __has_builtin(__builtin_amdgcn_tanhf)
    return __builtin_amdgcn_tanhf(x);        // v_tanh_f32 (trans pipe)
#else
    float e = __expf(-2.0f * x);
    return (1.0f - e) * __builtin_amdgcn_rcpf(1.0f + e);
#endif
}

__device__ __forceinline__ float fast_sigmoid(float x) {
#if __has_builtin(__builtin_amdgcn_tanhf)
    return fmaf(0.5f, __builtin_amdgcn_tanhf(0.5f * x), 0.5f);
#else
    float e = __expf(-x);
    return __builtin_amdgcn_rcpf(1.0f + e);
#endif
}

__global__ void lstm_zero_out(float* out) {
    if (threadIdx.x == 0) out[0] = 0.0f;
}

__global__ void __launch_bounds__(256)
lstm_proj_scan_kernel(const float* __restrict__ X,     // [B,T,P]
                      const float* __restrict__ x5,    // [B,P,K]
                      const float* __restrict__ W_ih,  // [4H,P]
                      const float* __restrict__ W_hh,  // [4H,P]
                      const float* __restrict__ b_ih,  // [4H]
                      const float* __restrict__ b_hh,  // [4H]
                      const float* __restrict__ W_hr,  // [P,H]
                      float* __restrict__ out) {
    const int lid  = threadIdx.x & 31;
    const int wave = threadIdx.x >> 5;
    const int b    = blockIdx.x * 8 + wave;
    if (b >= LSTM_B) return;

    const bool lo = (lid < 16);
    const int  n  = lid & 15;

    // ---- per-lane recurrent weights: rows {lid, 32+lid, 64+lid, 96+lid} ----
    float whh[4][3];
    float bsum[4];
#pragma unroll
    for (int g = 0; g < 4; ++g) {
        const int row = g * LSTM_H + lid;
        whh[g][0] = W_hh[row * 3 + 0];
        whh[g][1] = W_hh[row * 3 + 1];
        whh[g][2] = W_hh[row * 3 + 2];
        bsum[g]   = b_ih[row] + b_hh[row];
    }
    const float whr0 = W_hr[0 * LSTM_H + lid];
    const float whr1 = W_hr[1 * LSTM_H + lid];
    const float whr2 = W_hr[2 * LSTM_H + lid];

    // ---- B operands: W_ih^T tiles (4x16 f32 each), 8 tiles of 16 cols -----
    // VGPR0: lanes 0-15 = K0 row, lanes 16-31 = K2 row; VGPR1: K1 / K3(=0).
    v2f btv[8];
#pragma unroll
    for (int t = 0; t < 8; ++t) {
        const int col = t * 16 + n;
        btv[t].x = W_ih[col * 3 + (lo ? 0 : 2)];
        btv[t].y = lo ? W_ih[col * 3 + 1] : 0.0f;
    }

    float c_state = 0.0f;
    float h0 = 0.0f, h1 = 0.0f, h2 = 0.0f;

    const float* __restrict__ Xb = X + (size_t)b * LSTM_T * LSTM_P;

    // Branchless A-operand addressing for V_WMMA_F32_16X16X4_F32:
    // M = 16 timesteps, K = 4 (P=3 + zero pad).
    // lanes 0-15 (M=row): VGPR0=K0, VGPR1=K1; lanes 16-31: VGPR0=K2, VGPR1=0.
    // Per-lane row pointer and element offset computed ONCE; per block we do
    // two unconditional b32 loads (same cacheline) + one cndmask.
    const float* __restrict__ xlane = Xb + (size_t)n * LSTM_P;
    const int ofsA = lo ? 0 : 2;

    // ---- software pipeline: block 0 WMMAs up front ----
    v2f a_cur;
    {
        const float u = xlane[ofsA];
        const float w = xlane[1];
        a_cur.x = u;
        a_cur.y = lo ? w : 0.0f;
    }
    v8f d[8];
#pragma unroll
    for (int t = 0; t < 8; ++t) {
        v8f cz = {};
        d[t] = __builtin_amdgcn_wmma_f32_16x16x4_f32(
            false, a_cur, false, btv[t], (short)0, cz, false, false);
    }

    for (int blk = 0; blk < LSTM_T / 16; ++blk) {
        const int t0 = blk * 16;

        // Load next block's A early (prefetch one block beyond that).
        v2f a_next = a_cur;
        const bool have_next = (blk + 1 < LSTM_T / 16);
        if (have_next) {
            if (blk + 2 < LSTM_T / 16)
                __builtin_prefetch(Xb + (size_t)(t0 + 32) * LSTM_P, 0, 0);
            const float* xr = xlane + (size_t)(t0 + 16) * LSTM_P;
            const float u = xr[ofsA];
            const float w = xr[1];
            a_next.x = u;
            a_next.y = lo ? w : 0.0f;
        }

        // ---- steer all 16 steps' gate inputs out of the D tiles ----------
        // D layout: vgpr v: lanes 0-15 -> (M=v, N=lane), lanes 16-31 ->
        // (M=v+8, N=lane-16). One permlane16_swap of (tile 2g, tile 2g+1)
        // yields step v's distribution in x and step v+8's in y.
        float gxa[8][4], gxb[8][4];
#pragma unroll
        for (int v = 0; v < 8; ++v) {
#pragma unroll
            for (int g = 0; g < 4; ++g) {
                float x = d[2 * g + 0][v];
                float y = d[2 * g + 1][v];
                half_swap(x, y);
                gxa[v][g] = x;   // for step m = v
                gxb[v][g] = y;   // for step m = v + 8
            }
        }

        // ---- issue next block's WMMAs; they retire during the 16 steps ----
        if (have_next) {
#pragma unroll
            for (int t = 0; t < 8; ++t) {
                v8f cz = {};
                d[t] = __builtin_amdgcn_wmma_f32_16x16x4_f32(
                    false, a_next, false, btv[t], (short)0, cz, false, false);
            }
        }
        a_cur = a_next;

        // ---- 16 sequential LSTM steps -------------------------------------
#pragma unroll
        for (int m = 0; m < 16; ++m) {
            float gate[4];
#pragma unroll
            for (int g = 0; g < 4; ++g) {
                const float gx = (m < 8) ? gxa[m][g] : gxb[m - 8][g];
                float hp = fmaf(whh[g][0], h0, bsum[g]);
                hp       = fmaf(whh[g][1], h1, hp);
                hp       = fmaf(whh[g][2], h2, hp);
                gate[g]  = gx + hp;
            }

            const float ig = fast_sigmoid(gate[0]);
            const float fg = fast_sigmoid(gate[1]);
            const float gg = fast_tanh(gate[2]);
            const float og = fast_sigmoid(gate[3]);

            c_state = fmaf(fg, c_state, ig * gg);
            const float hid = og * fast_tanh(c_state);

            // h = hid @ W_hr^T : three all-lane weighted sums, VALU-only.
            h0 = wave_allsum(whr0 * hid);
            h1 = wave_allsum(whr1 * hid);
            h2 = wave_allsum(whr2 * hid);
        }
    }

    // ---- epilogue: score[b,k] = h . x5[b,:,k]; loss = softplus(-score) ----
    const float* __restrict__ x5b = x5 + (size_t)b * LSTM_P * LSTM_K;
    float acc = 0.0f;
#pragma unroll
    for (int kk = 0; kk < 4; ++kk) {
        const int k = lid + kk * 32;
        float s = h0 * x5b[k];
        s = fmaf(h1, x5b[LSTM_K + k], s);
        s = fmaf(h2, x5b[2 * LSTM_K + k], s);
        const float z  = -s;
        const float mx = fmaxf(z, 0.0f);
        acc += mx + __logf(__expf(z - mx) + __expf(-mx));
    }
    acc = wave_allsum(acc);

    if (lid == 0) {
        const float scale = 1.0f / ((float)LSTM_B * (float)LSTM_K);
        unsafeAtomicAdd(out, acc * scale);   // global_atomic_add_f32
    }
}

extern "C" void kernel_launch(void* const* d_in, const int* in_sizes, int n_in,
                              void* d_out, int out_size, void* d_ws, size_t ws_size,
                              hipStream_t stream) {
    (void)in_sizes; (void)n_in; (void)d_ws; (void)ws_size; (void)out_size;
    const float* X    = (const float*)d_in[0];
    const float* x5   = (const float*)d_in[1];
    const float* W_ih = (const float*)d_in[2];
    const float* W_hh = (const float*)d_in[3];
    const float* b_ih = (const float*)d_in[4];
    const float* b_hh = (const float*)d_in[5];
    const float* W_hr = (const float*)d_in[6];
    float* out = (float*)d_out;

    lstm_zero_out<<<1, 32, 0, stream>>>(out);
    // 8 waves per block, one wave per batch element: 4096/8 = 512 blocks.
    lstm_proj_scan_kernel<<<LSTM_B / 8, 256, 0, stream>>>(
        X, x5, W_ih, W_hh, b_ih, b_hh, W_hr, out);
}